// SBGNNLayer_68719476996
// MI455X (gfx1250) — compile-verified
//
#include <hip/hip_runtime.h>

typedef __attribute__((ext_vector_type(2))) float v2f;
typedef __attribute__((ext_vector_type(8))) float v8f;

#define NN 100000
#define EE 1000000
#define DD 64

// D = A(16x4 f32) * B(4x16 f32) + C(16x16 f32), wave32.
// 8 args: (neg_a, A, neg_b, B, c_mod, C, reuse_a, reuse_b)
__device__ __forceinline__ v8f wmma_f32_k4(v2f a, v2f b, v8f c) {
  return __builtin_amdgcn_wmma_f32_16x16x4_f32(false, a, false, b, (short)0, c,
                                               false, false);
}

// Stage NROWS x K row-major weights from global into LDS with padded leading
// dim LDP, using async global->LDS b128 copies (ASYNCcnt path, no VGPR
// staging). chunks is a multiple of 256 for all instantiations, so the loop
// is divergence-free. Ends with a full workgroup barrier.
template <int NROWS, int K, int LDP>
__device__ __forceinline__ void stage_w_lds(float* sdst,
                                            const float* __restrict__ gsrc) {
  constexpr int chunks = NROWS * (K / 4);  // 16B chunks
  const unsigned lds_base = (unsigned)(uintptr_t)sdst;  // low 32 = LDS offset
  for (int j = threadIdx.x; j < chunks; j += 256) {
    const int row = j / (K / 4);
    const int col = (j % (K / 4)) * 4;
    const unsigned loff = lds_base + (unsigned)(row * LDP + col) * 4u;
    const float* g = gsrc + (size_t)row * K + col;
    asm volatile("global_load_async_to_lds_b128 %0, %1, off"
                 :: "v"(loff), "v"(g)
                 : "memory");
  }
  asm volatile("s_wait_asynccnt 0" ::: "memory");
  __syncthreads();
}

// acc[NT] (16 rows x NT*16 cols) += Atile(16x64, global) * Wlds^T.
// Wlds: LDS row-major [N x ldw] (padded), B[k][n] = W[n][k].
template <int NT>
__device__ __forceinline__ void accum_k64(const float* __restrict__ A, int lda,
                                          const float* Wlds, int ldw, int lane,
                                          v8f* acc) {
  const int row = lane & 15;        // M (for A) / N (for B) within tile
  const int kh = (lane >> 4) * 2;   // K sub-offset: lanes 0-15 -> 0, 16-31 -> 2
  const float* arow = A + (size_t)row * lda;
#pragma unroll 4
  for (int k0 = 0; k0 < 64; k0 += 4) {
    v2f a = *(const v2f*)(arow + k0 + kh);
#pragma unroll
    for (int nt = 0; nt < NT; ++nt) {
      v2f b = *(const v2f*)(Wlds + (nt * 16 + row) * ldw + k0 + kh);
      acc[nt] = wmma_f32_k4(a, b, acc[nt]);
    }
  }
}

// msg = feat @ W^T + bias   (M x 64) <- (M x 64)(64 x 64)
__global__ __launch_bounds__(256) void transform_kernel(
    const float* __restrict__ feat, const float* __restrict__ W,
    const float* __restrict__ bias, float* __restrict__ out, int M) {
  __shared__ float sW[64 * 68];  // pad 64 -> 68 (= 4 mod 64 banks)
  stage_w_lds<64, 64, 68>(sW, W);

  const int lane = threadIdx.x & 31;
  const int wave = blockIdx.x * 8 + (threadIdx.x >> 5);
  const int m0 = wave * 16;
  if (m0 < M) {  // wave-uniform: EXEC all-ones for WMMA
    v8f acc[4] = {};
    accum_k64<4>(feat + (size_t)m0 * 64, 64, sW, 68, lane, acc);
    const int row = lane & 15;
    const int mh = m0 + (lane >> 4) * 8;
#pragma unroll
    for (int nt = 0; nt < 4; ++nt) {
      const int n = nt * 16 + row;
      const float bn = bias[n];
#pragma unroll
      for (int r = 0; r < 8; ++r)
        out[(size_t)(mh + r) * 64 + n] = acc[nt][r] + bn;
    }
  }
}

__global__ void zero_kernel(float4* __restrict__ p, int n4) {
  int t = blockIdx.x * blockDim.x + threadIdx.x;
  if (t < n4) p[t] = make_float4(0.f, 0.f, 0.f, 0.f);
}

// For each edge: agg[dst] += msg[src]; deg[dst] += 1. 16 lanes per edge.
// msg + agg working set (~51 MB) is L2-resident (192 MB), so the random
// gather + fp32 atomics run at L2 rather than HBM speed.
__global__ void scatter_kernel(const int* __restrict__ edges,
                               const float* __restrict__ msg,
                               float* __restrict__ agg, float* __restrict__ deg,
                               int E) {
  int t = blockIdx.x * blockDim.x + threadIdx.x;
  int e = t >> 4;
  if (e >= E) return;
  int part = t & 15;
  // stream-prefetch edge indices well ahead (speculative: OOB is dropped)
  __builtin_prefetch(edges + 2 * e + 8192, 0, 0);
  int dst = edges[2 * e + 0];
  int src = edges[2 * e + 1];
  const float4 v = *(const float4*)(msg + (size_t)src * 64 + part * 4);
  float* ap = agg + (size_t)dst * 64 + part * 4;
  atomicAdd(ap + 0, v.x);
  atomicAdd(ap + 1, v.y);
  atomicAdd(ap + 2, v.z);
  atomicAdd(ap + 3, v.w);
  if (part == 0) atomicAdd(deg + dst, 1.0f);
}

// agg[row] *= 1 / max(deg[row], 1)
__global__ void normalize_kernel(float* __restrict__ agg,
                                 const float* __restrict__ deg, int nodes) {
  int t = blockIdx.x * blockDim.x + threadIdx.x;
  if (t >= nodes * DD) return;
  float d = deg[t >> 6];
  agg[t] *= 1.0f / fmaxf(d, 1.0f);
}

// h = prelu(concat([feat, ma, mb, mc, md]) @ W1^T + b1)  (M x 128), K = 320
// W1 staged to LDS in two 64-row halves (83 KB each, padded stride 324).
__global__ __launch_bounds__(256) void mlp1_kernel(
    const float* __restrict__ feat, const float* __restrict__ ma,
    const float* __restrict__ mb, const float* __restrict__ mc,
    const float* __restrict__ md, const float* __restrict__ W1,
    const float* __restrict__ b1, const float* __restrict__ prelu,
    float* __restrict__ h, int M) {
  __shared__ float sW[64 * 324];  // pad 320 -> 324 (= 4 mod 64 banks)

  const int lane = threadIdx.x & 31;
  const int wave = blockIdx.x * 8 + (threadIdx.x >> 5);
  const int m0 = wave * 16;
  const bool active = m0 < M;  // wave-uniform

  v8f acc[8] = {};
  const float* segs[5];
  segs[0] = feat; segs[1] = ma; segs[2] = mb; segs[3] = mc; segs[4] = md;

  // First half: output cols 0..63 -> acc[0..3]
  stage_w_lds<64, 320, 324>(sW, W1);
  if (active) {
#pragma unroll
    for (int s = 0; s < 5; ++s)
      accum_k64<4>(segs[s] + (size_t)m0 * 64, 64, sW + s * 64, 324, lane,
                   acc + 0);
  }
  __syncthreads();  // all LDS reads done before restaging

  // Second half: output cols 64..127 -> acc[4..7]
  stage_w_lds<64, 320, 324>(sW, W1 + (size_t)64 * 320);
  if (active) {
#pragma unroll
    for (int s = 0; s < 5; ++s)
      accum_k64<4>(segs[s] + (size_t)m0 * 64, 64, sW + s * 64, 324, lane,
                   acc + 4);

    const float pa = *prelu;
    const int row = lane & 15;
    const int mh = m0 + (lane >> 4) * 8;
#pragma unroll
    for (int nt = 0; nt < 8; ++nt) {
      const int n = nt * 16 + row;
      const float bn = b1[n];
#pragma unroll
      for (int r = 0; r < 8; ++r) {
        float v = acc[nt][r] + bn;
        h[(size_t)(mh + r) * 128 + n] = v > 0.f ? v : pa * v;
      }
    }
  }
}

// out = h @ W2^T + b2   (M x 64), K = 128
__global__ __launch_bounds__(256) void mlp2_kernel(
    const float* __restrict__ h, const float* __restrict__ W2,
    const float* __restrict__ b2, float* __restrict__ out, int M) {
  __shared__ float sW[64 * 132];  // pad 128 -> 132 (= 4 mod 64 banks)
  stage_w_lds<64, 128, 132>(sW, W2);

  const int lane = threadIdx.x & 31;
  const int wave = blockIdx.x * 8 + (threadIdx.x >> 5);
  const int m0 = wave * 16;
  if (m0 < M) {
    v8f acc[4] = {};
    accum_k64<4>(h + (size_t)m0 * 128 + 0, 128, sW + 0, 132, lane, acc);
    accum_k64<4>(h + (size_t)m0 * 128 + 64, 128, sW + 64, 132, lane, acc);
    const int row = lane & 15;
    const int mh = m0 + (lane >> 4) * 8;
#pragma unroll
    for (int nt = 0; nt < 4; ++nt) {
      const int n = nt * 16 + row;
      const float bn = b2[n];
#pragma unroll
      for (int r = 0; r < 8; ++r)
        out[(size_t)(mh + r) * 64 + n] = acc[nt][r] + bn;
    }
  }
}

extern "C" void kernel_launch(void* const* d_in, const int* in_sizes, int n_in,
                              void* d_out, int out_size, void* d_ws,
                              size_t ws_size, hipStream_t stream) {
  const float* featA = (const float*)d_in[0];
  const float* featB = (const float*)d_in[1];
  const float* W1 = (const float*)d_in[26];
  const float* b1 = (const float*)d_in[27];
  const float* prelu = (const float*)d_in[28];
  const float* W2 = (const float*)d_in[29];
  const float* b2 = (const float*)d_in[30];

  float* ws = (float*)d_ws;
  float* msg = ws;                          // NN*64
  float* agg = msg + (size_t)NN * DD;       // 4 * NN*64
  float* deg = agg + 4 * (size_t)NN * DD;   // 4 * NN
  float* h = deg + 4 * (size_t)NN;          // NN*128
  float* out = (float*)d_out;

  const int tileBlocks = (NN / 16 + 7) / 8;  // 782 blocks of 8 waves
  const dim3 blk(256);

  auto run_set = [&](const float* self, const int* tids,
                     const float* const* srcs, float* outp) {
    for (int t = 0; t < 4; ++t) {
      const int i = tids[t];
      const int* edges = (const int*)d_in[2 + i];
      const float* W = (const float*)d_in[10 + 2 * i];
      const float* b = (const float*)d_in[11 + 2 * i];
      float* aggT = agg + (size_t)t * NN * DD;
      float* degT = deg + (size_t)t * NN;
      transform_kernel<<<tileBlocks, blk, 0, stream>>>(srcs[t], W, b, msg, NN);
      zero_kernel<<<(NN * DD / 4 + 255) / 256, blk, 0, stream>>>((float4*)aggT,
                                                                 NN * DD / 4);
      zero_kernel<<<(NN / 4 + 255) / 256, blk, 0, stream>>>((float4*)degT,
                                                            NN / 4);
      scatter_kernel<<<(EE * 16) / 256, blk, 0, stream>>>(edges, msg, aggT,
                                                          degT, EE);
      normalize_kernel<<<(NN * DD + 255) / 256, blk, 0, stream>>>(aggT, degT,
                                                                  NN);
    }
    mlp1_kernel<<<tileBlocks, blk, 0, stream>>>(
        self, agg, agg + (size_t)NN * DD, agg + 2 * (size_t)NN * DD,
        agg + 3 * (size_t)NN * DD, W1, b1, prelu, h, NN);
    mlp2_kernel<<<tileBlocks, blk, 0, stream>>>(h, W2, b2, outp, NN);
  };

  // Set A: m_ab_pos, m_ab_neg (src = feature_b); m_aa_pos, m_aa_neg (src = a)
  const int tidsA[4] = {0, 1, 4, 5};
  const float* srcA[4] = {featB, featB, featA, featA};
  run_set(featA, tidsA, srcA, out);

  // Set B: m_ba_pos, m_ba_neg (src = feature_a); m_bb_pos, m_bb_neg (src = b)
  const int tidsB[4] = {2, 3, 6, 7};
  const float* srcB[4] = {featA, featA, featB, featB};
  run_set(featB, tidsB, srcB, out + (size_t)NN * DD);
}